// DPCE_loss_15169824489490
// MI455X (gfx1250) — compile-verified
//
#include <hip/hip_runtime.h>
#include <hip/hip_bf16.h>

// Problem geometry (fixed by the reference)
#define SPX   4718592            // D*H*W = 128*192*192
#define NBATCH 2
#define NCH    4
#define NPTS  (NBATCH * SPX)     // 9,437,184
#define EPSF  1e-8f
#define NBLK  1536
#define TPB   256

typedef __attribute__((ext_vector_type(2))) float v2f;
typedef __attribute__((ext_vector_type(8))) float v8f;

// ---------------------------------------------------------------------------
// Wave-level (wave32) sum reduction using V_WMMA_F32_16X16X4_F32.
// A(16x4): lane L<16 supplies K=0 (vgpr0) / K=1 (vgpr1); lane L>=16 supplies
// K=2/K=3.  With A = {p, 0} per lane and B = ones, D[m][n] = p_m + p_{m+16}.
// C/D layout: lane n<16 holds D[r][n] in vgpr r; lanes 16-31 hold D[8+r][..].
// Sum of the 8 result VGPRs per lane = half-wave total; one xor-16 shuffle
// completes the full 32-lane sum.  Fixed HW accumulation order => determinism.
// Requires EXEC == all ones (call unconditionally, outside divergent flow).
// ---------------------------------------------------------------------------
__device__ __forceinline__ float wave_sum_wmma(float p) {
  v2f a; a[0] = p;    a[1] = 0.0f;
  v2f b; b[0] = 1.0f; b[1] = 1.0f;
  v8f c = {};
  c = __builtin_amdgcn_wmma_f32_16x16x4_f32(
      /*neg_a=*/false, a, /*neg_b=*/false, b,
      /*c_mod=*/(short)0, c, /*reuse_a=*/false, /*reuse_b=*/false);
  float s = ((c[0] + c[1]) + (c[2] + c[3])) + ((c[4] + c[5]) + (c[6] + c[7]));
  s += __shfl_xor(s, 16, 32);
  return s;   // identical in every lane
}

// ---------------------------------------------------------------------------
// ws layout (uints/floats): [0..5] per-(b,c) min bits, [6..11] max bits,
// [16..16+NBLK) block partial sums (float).
// ---------------------------------------------------------------------------
__global__ void k_init(unsigned* __restrict__ mm) {
  int t = threadIdx.x;
  if (t < 6)       mm[t] = 0x7F800000u;  // +inf  (min identity)
  else if (t < 12) mm[t] = 0x00000000u;  // 0.0f  (max identity; all z >= 0)
}

__device__ __forceinline__ void upd3(float (&mn)[3], float (&mx)[3], int tk, float dk) {
#pragma unroll
  for (int c = 1; c < 4; ++c) {
    float v = (tk == c) ? dk : 0.0f;
    mn[c - 1] = fminf(mn[c - 1], v);
    mx[c - 1] = fmaxf(mx[c - 1], v);
  }
}

__global__ void k_minmax(const int* __restrict__ tgt2,      // int64 viewed as int pairs
                         const float* __restrict__ dist,
                         unsigned* __restrict__ mm) {
  const float INF = __int_as_float(0x7F800000);
  float mnA[3] = {INF, INF, INF}, mxA[3] = {0.f, 0.f, 0.f};   // b = 0
  float mnB[3] = {INF, INF, INF}, mxB[3] = {0.f, 0.f, 0.f};   // b = 1

  const int stride = NBLK * TPB * 4;
  for (int i4 = (blockIdx.x * TPB + threadIdx.x) * 4; i4 < NPTS; i4 += stride) {
    const int  b  = (i4 >= SPX) ? 1 : 0;           // uniform per wave (SPX % 128 == 0)
    float4 d4 = ((const float4*)dist)[i4 >> 2];
    int4   ta = ((const int4*)tgt2)[i4 >> 1];      // points i4, i4+1 (low words)
    int4   tb = ((const int4*)tgt2)[(i4 >> 1) + 1];
    int   t[4]  = {ta.x, ta.z, tb.x, tb.z};
    float dv[4] = {d4.x, d4.y, d4.z, d4.w};
    if (b == 0) {
#pragma unroll
      for (int k = 0; k < 4; ++k) upd3(mnA, mxA, t[k], dv[k]);
    } else {
#pragma unroll
      for (int k = 0; k < 4; ++k) upd3(mnB, mxB, t[k], dv[k]);
    }
  }

#pragma unroll
  for (int b = 0; b < 2; ++b) {
#pragma unroll
    for (int c = 0; c < 3; ++c) {
      float v0 = b ? mnB[c] : mnA[c];
      float v1 = b ? mxB[c] : mxA[c];
      for (int off = 16; off > 0; off >>= 1) {
        v0 = fminf(v0, __shfl_xor(v0, off, 32));
        v1 = fmaxf(v1, __shfl_xor(v1, off, 32));
      }
      if ((threadIdx.x & 31) == 0) {               // nonneg floats: uint order == float order
        atomicMin(&mm[b * 3 + c],     __float_as_uint(v0));
        atomicMax(&mm[6 + b * 3 + c], __float_as_uint(v1));
      }
    }
  }
}

__device__ __forceinline__ float point_contrib(float a0, float a1, float a2, float a3,
                                               int tk, float dk,
                                               const float (&mn)[3], const float (&inv)[3]) {
  float m   = fmaxf(fmaxf(a0, a1), fmaxf(a2, a3));
  float e   = __expf(a0 - m) + __expf(a1 - m) + __expf(a2 - m) + __expf(a3 - m);
  float lse = m + __logf(e);
  float pt  = (tk == 0) ? (a0 - lse) : 0.0f;
  float z1  = ((tk == 1 ? dk : 0.0f) - mn[0]) * inv[0];
  float z2  = ((tk == 2 ? dk : 0.0f) - mn[1]) * inv[1];
  float z3  = ((tk == 3 ? dk : 0.0f) - mn[2]) * inv[2];
  pt += z1 * (a1 - lse) + z2 * (a2 - lse) + z3 * (a3 - lse);
  return pt;
}

__global__ void k_loss(const float* __restrict__ net,
                       const int* __restrict__ tgt2,
                       const float* __restrict__ dist,
                       const unsigned* __restrict__ mm,
                       float* __restrict__ partial) {
  __shared__ float wsum8[TPB / 32];

  float mn0[3], iv0[3], mn1[3], iv1[3];
#pragma unroll
  for (int c = 0; c < 3; ++c) {
    float a = __uint_as_float(mm[c]),     A = __uint_as_float(mm[6 + c]);
    float b = __uint_as_float(mm[3 + c]), B = __uint_as_float(mm[9 + c]);
    mn0[c] = a; iv0[c] = 1.0f / (A + EPSF - a);
    mn1[c] = b; iv1[c] = 1.0f / (B + EPSF - b);
  }

  float acc = 0.0f;
  const int stride = NBLK * TPB * 4;    // NPTS / stride == 6 exactly (uniform trips)
  for (int i4 = (blockIdx.x * TPB + threadIdx.x) * 4; i4 < NPTS; i4 += stride) {
    const int b = (i4 >= SPX) ? 1 : 0;
    const int s = i4 - b * SPX;
    const float* base = net + (size_t)b * NCH * SPX + s;

    float4 xc[4];
#pragma unroll
    for (int c = 0; c < 4; ++c) xc[c] = *(const float4*)(base + (size_t)c * SPX);
    float4 d4 = ((const float4*)dist)[i4 >> 2];
    int4   ta = ((const int4*)tgt2)[i4 >> 1];
    int4   tb = ((const int4*)tgt2)[(i4 >> 1) + 1];
    int   t[4]  = {ta.x, ta.z, tb.x, tb.z};
    float dv[4] = {d4.x, d4.y, d4.z, d4.w};

#pragma unroll
    for (int k = 0; k < 4; ++k) {
      float a0 = ((const float*)&xc[0])[k];
      float a1 = ((const float*)&xc[1])[k];
      float a2 = ((const float*)&xc[2])[k];
      float a3 = ((const float*)&xc[3])[k];
      acc += (b == 0) ? point_contrib(a0, a1, a2, a3, t[k], dv[k], mn0, iv0)
                      : point_contrib(a0, a1, a2, a3, t[k], dv[k], mn1, iv1);
    }
  }

  // Unconditional, post-loop: EXEC is all ones here (wave32, full blocks).
  float ws = wave_sum_wmma(acc);
  if ((threadIdx.x & 31) == 0) wsum8[threadIdx.x >> 5] = ws;
  __syncthreads();
  if (threadIdx.x == 0) {
    float bsum = 0.0f;
#pragma unroll
    for (int w = 0; w < TPB / 32; ++w) bsum += wsum8[w];
    partial[blockIdx.x] = bsum;
  }
}

__global__ void k_final(const float* __restrict__ partial, float* __restrict__ out) {
  __shared__ double sh[TPB];
  double s = 0.0;
  for (int i = threadIdx.x; i < NBLK; i += TPB) s += (double)partial[i];
  sh[threadIdx.x] = s;
  __syncthreads();
  for (int off = TPB / 2; off > 0; off >>= 1) {   // fixed pairing -> deterministic
    if (threadIdx.x < off) sh[threadIdx.x] += sh[threadIdx.x + off];
    __syncthreads();
  }
  if (threadIdx.x == 0) out[0] = (float)(-sh[0] / (double)NPTS);
}

extern "C" void kernel_launch(void* const* d_in, const int* in_sizes, int n_in,
                              void* d_out, int out_size, void* d_ws, size_t ws_size,
                              hipStream_t stream) {
  const float* net  = (const float*)d_in[0];   // [2,4,128,192,192] f32
  const int*   tgt2 = (const int*)d_in[1];     // [2,1,128,192,192] int64 -> int pairs
  const float* dist = (const float*)d_in[2];   // [2,128,192,192]   f32

  unsigned* mm      = (unsigned*)d_ws;         // 12 min/max slots
  float*    partial = (float*)d_ws + 16;       // NBLK block partials

  k_init  <<<1,    32,  0, stream>>>(mm);
  k_minmax<<<NBLK, TPB, 0, stream>>>(tgt2, dist, mm);
  k_loss  <<<NBLK, TPB, 0, stream>>>(net, tgt2, dist, mm, partial);
  k_final <<<1,    TPB, 0, stream>>>(partial, (float*)d_out);
}